// Confidence_Loss_33131377721741
// MI455X (gfx1250) — compile-verified
//
#include <hip/hip_runtime.h>
#include <hip/hip_bf16.h>

// Confidence loss, B=16, H=W=1024 (shapes per reference setup_inputs).
// Memory-bound streaming reduction: ~470MB traffic -> ~20us roofline @23.3TB/s.
// Main kernel: float4 streaming of the 3 o_f planes, low-dword-only reads of
// the int64 target, v_log_f32 fast log (accumulate log2, scale by -ln2 once),
// deterministic wave32-shuffle + LDS tree block reduction.
// Finalize kernel: one wave; 32 partials -> scalar entirely on the matrix pipe
// via two V_WMMA_F32_16X16X4_F32 multiplications by an all-ones B matrix.

#define HW_SHIFT 20              // H*W = 1<<20
#define W_SHIFT 10               // W   = 1<<10
#define W_MASK 1023

typedef __attribute__((ext_vector_type(2))) float v2f;
typedef __attribute__((ext_vector_type(8))) float v8f;

__global__ void __launch_bounds__(256)
conf_loss_main(const float* __restrict__ of,
               const int* __restrict__ tgt_lo,   // int64 target viewed as dword pairs (LE: [2*i] is low word)
               float* __restrict__ partials,
               int n_chunks)                      // n_pixels / 4
{
    const int HW = 1 << HW_SHIFT;
    float acc = 0.0f;   // accumulates log2(arg); scaled by -ln2 at the end

    for (int c = blockIdx.x * blockDim.x + threadIdx.x; c < n_chunks;
         c += gridDim.x * blockDim.x) {
        const int p0 = c << 2;                       // first pixel of chunk
        const int b  = p0 >> HW_SHIFT;
        const int within = p0 & (HW - 1);
        const int row = within >> W_SHIFT;
        const int col0 = within & W_MASK;            // chunk never crosses a row (W%4==0)

        const size_t base = (size_t)b * (3u << HW_SHIFT) + (size_t)within;
        const float4 o0 = *(const float4*)(of + base);            // x-offset plane
        const float4 o1 = *(const float4*)(of + base + HW);       // y-offset plane
        const float4 o2 = *(const float4*)(of + base + 2 * HW);   // confidence plane

        // low dwords of 4 consecutive int64 targets: dword indices 2*p0 .. 2*p0+7
        const int4 ta = *(const int4*)(tgt_lo + ((size_t)p0 << 1));
        const int4 tb = *(const int4*)(tgt_lo + ((size_t)p0 << 1) + 4);

        const float ox[4] = {o0.x, o0.y, o0.z, o0.w};
        const float oy[4] = {o1.x, o1.y, o1.z, o1.w};
        const float fc[4] = {o2.x, o2.y, o2.z, o2.w};
        const int   tg[4] = {ta.x, ta.z, tb.x, tb.z};

        const float rowf = (float)row;
#pragma unroll
        for (int k = 0; k < 4; ++k) {
            int x = __float2int_rd(ox[k] + (float)(col0 + k));
            int y = __float2int_rd(oy[k] + rowf);
            x = min(max(x, 0), W_MASK);
            y = min(max(y, 0), W_MASK);
            int tt = tg[k];            // int64 value fits in low dword; -1 stays -1
            tt = (tt == -1) ? 0 : tt;
            const size_t g = ((size_t)b << HW_SHIFT) + ((size_t)y << W_SHIFT) + (size_t)x;
            int hs = tgt_lo[g << 1];   // gather low dword
            hs = (hs == -1) ? 0 : hs;
            const float f = fc[k];
            const float arg = (tt == hs) ? (f + 1e-7f) : (1.0f - f + 1e-7f);
            acc += __log2f(arg);       // v_log_f32
        }
    }

    acc *= -0.69314718055994530942f;   // -ln(2): convert sum(log2) -> sum(-ln)

    // wave32 butterfly reduction (deterministic)
#pragma unroll
    for (int m = 16; m >= 1; m >>= 1)
        acc += __shfl_xor(acc, m, 32);

    __shared__ float warp_sums[8];     // 256 threads = 8 waves
    const int wave = threadIdx.x >> 5;
    if ((threadIdx.x & 31) == 0) warp_sums[wave] = acc;
    __syncthreads();
    if (threadIdx.x == 0) {
        float s = 0.0f;
#pragma unroll
        for (int i = 0; i < 8; ++i) s += warp_sums[i];   // fixed order: deterministic
        partials[blockIdx.x] = s;
    }
}

// One wave. Full 32-lane -> scalar reduction on the matrix pipe:
//   WMMA#1: A(16x4) holds p_L in K=0 (lanes 0-15 supply M=L, lanes 16-31 M=L-16
//           at K=2), B = ones, C = 0  =>  every D element of row M equals
//           s_M = p_M + p_{M+16}; lane L<16 holds s_0..s_7 in its 8 D VGPRs,
//           lanes >=16 hold s_8..s_15.
//   Sum the 8 D VGPRs per lane  =>  lanes 0-15: U_lo, lanes 16-31: U_hi.
//   WMMA#2: same ones-trick on U  =>  every D element = U_lo + U_hi = total.
__global__ void __launch_bounds__(32)
conf_loss_finalize(const float* __restrict__ partials, int n, float scale,
                   float* __restrict__ out)
{
    float p = 0.0f;
    for (int i = threadIdx.x; i < n; i += 32) p += partials[i];

    v2f ones; ones[0] = 1.0f; ones[1] = 1.0f;
    v8f c = {};

    v2f a1; a1[0] = p; a1[1] = 0.0f;
    v8f d1 = __builtin_amdgcn_wmma_f32_16x16x4_f32(
        false, a1, false, ones, (short)0, c, false, false);

    float u = 0.0f;
#pragma unroll
    for (int v = 0; v < 8; ++v) u += d1[v];

    v2f a2; a2[0] = u; a2[1] = 0.0f;
    v8f d2 = __builtin_amdgcn_wmma_f32_16x16x4_f32(
        false, a2, false, ones, (short)0, c, false, false);

    if (threadIdx.x == 0) out[0] = d2[0] * scale;
}

extern "C" void kernel_launch(void* const* d_in, const int* in_sizes, int n_in,
                              void* d_out, int out_size, void* d_ws, size_t ws_size,
                              hipStream_t stream) {
    const float* of     = (const float*)d_in[0];   // (B,3,H,W) float32
    const int*   tgt_lo = (const int*)d_in[1];     // (B,H,W) int64 -> dword pairs
    float*       out    = (float*)d_out;           // scalar float32
    float*       parts  = (float*)d_ws;

    const int n_pix    = in_sizes[1];              // B*H*W
    const int n_chunks = n_pix >> 2;

    int nb = 4096;
    const int max_nb = (n_chunks + 255) / 256;
    if (nb > max_nb) nb = max_nb;

    conf_loss_main<<<nb, 256, 0, stream>>>(of, tgt_lo, parts, n_chunks);

    const float scale = 1.0f / (float)n_pix;       // W_F * mean(batch sums / HW)
    conf_loss_finalize<<<1, 32, 0, stream>>>(parts, nb, scale, out);
}